// MACL_82394652607212
// MI455X (gfx1250) — compile-verified
//
#include <hip/hip_runtime.h>
#include <hip/hip_bf16.h>
#include <math.h>
#include <stdint.h>

// MACL supervised-contrastive loss, fully fused on gfx1250 (MI455X).
// B=1024, V=4, D=512, N=4096.  cs = X X^T via V_WMMA_F32_16X16X4_F32.
// gfx1250 paths: f32 WMMA matrix pipe + GLOBAL_LOAD_ASYNC_TO_LDS staging with
// an 8-slot / 6-chunk-lookahead ring (ASYNCcnt <= 12 in flight), relaxed WAR
// guard (s_wait_dscnt 3) so LDS fragment loads stay pipelined across chunks,
// large-LDS (130 KB/WG), masks + online LSE fused, deterministic reduction.

typedef __attribute__((ext_vector_type(2))) float v2f;
typedef __attribute__((ext_vector_type(8))) float v8f;

#define TEMP_INV (1.0f / 0.07f)
#define NNEG (-1.0e30f)

// ---- gfx1250 async memory->LDS helpers (per-lane addressing) -------------
__device__ __forceinline__ void async_lds_b128(unsigned lds_addr, const float* g) {
    asm volatile("global_load_async_to_lds_b128 %0, %1, off"
                 :: "v"(lds_addr), "v"(g) : "memory");
}
__device__ __forceinline__ void async_lds_b64(unsigned lds_addr, const float* g) {
    asm volatile("global_load_async_to_lds_b64 %0, %1, off"
                 :: "v"(lds_addr), "v"(g) : "memory");
}
__device__ __forceinline__ void wait_async_all() { asm volatile("s_wait_asynccnt 0"  ::: "memory"); }
__device__ __forceinline__ void wait_async_12()  { asm volatile("s_wait_asynccnt 12" ::: "memory"); }
__device__ __forceinline__ void wait_ds_all()    { asm volatile("s_wait_dscnt 0"     ::: "memory"); }
// all but the newest chunk's (>=3-instruction) LDS loads have returned
__device__ __forceinline__ void wait_ds_soft()   { asm volatile("s_wait_dscnt 3"     ::: "memory"); }

// Block: 16 rows of the Gram matrix; 8 waves each own 16 tile-pairs (2 column
// tiles per pass -> A-fragment reuse).  A strip async-staged pair-interleaved
// [k/2][row][k%2] (conflict-free b64 frags); B chunks (16 cols x 8 k per tile)
// async-staged into an 8-slot ring of stride-12 row-major slabs.
__global__ __launch_bounds__(256) void macl_fused_main(
    const float* __restrict__ feats,   // [1024][4][512]
    const int* __restrict__ labels,    // [1024]
    float* __restrict__ partial)       // [256] per-block partials
{
    __shared__ __align__(16) float A_pi[16 * 512];     // 32 KB
    __shared__ __align__(16) float B_db[8 * 8 * 384];  // 96 KB: wave x slot x tile x (16x12)
    __shared__ float red[8][16][5];

    const int tid = threadIdx.x;
    const int wv  = tid >> 5;
    const int ln  = tid & 31;
    const int hi  = ln >> 4;           // half-wave select (WMMA operand layout)
    const int nl  = ln & 15;
    const int rowbase = blockIdx.x * 16;
    const int rblk  = rowbase >> 10;   // 4x4 block-row (uniform)
    const int abase = rowbase & 1023;  // batch-index base (uniform)

    // ---- async-stage A strip: contrast row i -> feature row (i&1023)*4+(i>>10)
    {
        const int r = tid & 15;
        const long frow = (long)((abase + r) * 4 + rblk) * 512;
        const unsigned lbase = (unsigned)(uintptr_t)(&A_pi[0]);
        #pragma unroll
        for (int q = 0; q < 16; ++q) {
            const int p = (tid >> 4) + 16 * q;     // k-pair index 0..255
            async_lds_b64(lbase + (unsigned)((p * 32 + 2 * r) * 4),
                          feats + frow + 2 * p);
        }
        wait_async_all();
    }
    __syncthreads();

    // per-lane online stats for 8 rows (row = v + 8*hi of the strip)
    float m[8], sE[8], S1[8], S2[8], Cn[8];
    int la[8];
    #pragma unroll
    for (int v = 0; v < 8; ++v) {
        m[v] = NNEG; sE[v] = 0.f; S1[v] = 0.f; S2[v] = 0.f; Cn[v] = 0.f;
        la[v] = labels[abase + v + 8 * hi];
    }

    const float*   Bw    = &B_db[wv * 3072];
    const unsigned Bbase = (unsigned)(uintptr_t)Bw;
    const unsigned laneB = (unsigned)((nl * 12 + 4 * hi) * 4); // lane slot in tile slab

    for (int gi = 0; gi < 16; ++gi) {
        const int pr      = wv + 8 * gi;     // tile-pair 0..127 (32 cols)
        const int colbase = pr * 32;
        const int cblk    = colbase >> 10;
        const int bbase   = colbase & 1023;
        const int idx     = (rblk - cblk) & 3;

        // per-lane global pointers to this lane's k-quad of each tile's column
        const float* g0 = feats + (long)((bbase + nl)      * 4 + cblk) * 512 + 4 * hi;
        const float* g1 = feats + (long)((bbase + 16 + nl) * 4 + cblk) * 512 + 4 * hi;

        v8f acc0 = {0.f, 0.f, 0.f, 0.f, 0.f, 0.f, 0.f, 0.f};
        v8f acc1 = {0.f, 0.f, 0.f, 0.f, 0.f, 0.f, 0.f, 0.f};

        auto issue = [&](int chunk, int slot) {
            const unsigned bo = (unsigned)(slot * 384 * 4);
            async_lds_b128(Bbase + bo + laneB,           g0 + chunk * 8);
            async_lds_b128(Bbase + bo + 192 * 4 + laneB, g1 + chunk * 8);
        };
        auto compute = [&](int kc, int bufo) {
            #pragma unroll
            for (int k8 = 0; k8 < 8; k8 += 4) {
                v2f af = *(const v2f*)(&A_pi[((kc + k8) >> 1) * 32 + hi * 32 + 2 * nl]);
                v2f b0 = *(const v2f*)(&Bw[bufo +       nl * 12 + k8 + 2 * hi]);
                v2f b1 = *(const v2f*)(&Bw[bufo + 192 + nl * 12 + k8 + 2 * hi]);
                acc0 = __builtin_amdgcn_wmma_f32_16x16x4_f32(
                    false, af, false, b0, (short)0, acc0, false, false);
                acc1 = __builtin_amdgcn_wmma_f32_16x16x4_f32(
                    false, af, false, b1, (short)0, acc1, false, false);
            }
        };

        // prologue: fill 6 of 8 ring slots (12 asyncs in flight)
        wait_ds_all();                       // prior pass's ring reads complete
        #pragma unroll
        for (int d = 0; d < 6; ++d) issue(d, d);

        // steady state: 6-chunk lookahead; refill target was read 2 chunks ago
        for (int c = 0; c < 58; ++c) {
            wait_ds_soft();                  // WAR: only newest chunk's reads may be in flight
            issue(c + 6, (c + 6) & 7);
            wait_async_12();                 // chunk c landed (in-order completion)
            compute(c * 8, (c & 7) * 384);
        }
        // tail: everything already issued; one drain, then compute chunks 58..63
        wait_async_all();
        #pragma unroll
        for (int c = 58; c < 64; ++c) compute(c * 8, (c & 7) * 384);

        // ---- epilogue: masks + online LSE for both 16x16 tiles
        const int   lb0 = labels[bbase + nl];
        const int   lb1 = labels[bbase + 16 + nl];
        const float wgt = (idx == 2) ? 1.25f : 1.0f;
        #pragma unroll
        for (int ti = 0; ti < 2; ++ti) {
            const int lb   = ti ? lb1 : lb0;
            const int bcol = bbase + 16 * ti + nl;
            #pragma unroll
            for (int v = 0; v < 8; ++v) {
                const float cs  = ti ? acc1[v] : acc0[v];
                const int aRow  = abase + v + 8 * hi;
                const bool eq   = (la[v] == lb);
                const bool negm = ((idx & 1) == 0) && !eq;
                const float th  = tanhf(cs);
                const float adc = negm ? (4.0f * th * TEMP_INV) : (cs * TEMP_INV);
                const bool ab   = (aRow == bcol);
                const bool fm   = ab || ((idx == 2) && eq);
                const bool diag = (rblk == cblk) && ab;
                const bool pm   = fm && !diag;
                const float p   = pm ? wgt : 0.0f;
                S1[v] += p * adc;
                S2[v] += p;
                Cn[v] += fm ? 1.0f : 0.0f;
                const float e  = diag ? NNEG : adc;
                const float nm = fmaxf(m[v], e);
                sE[v] = sE[v] * __expf(m[v] - nm) + __expf(e - nm);
                m[v]  = nm;
            }
        }
    }

    // ---- cross-lane reduce (16 lanes share each row within a half-wave)
    #pragma unroll
    for (int v = 0; v < 8; ++v) {
        for (int off = 1; off < 16; off <<= 1) {
            const float mo = __shfl_xor(m[v], off, 32);
            const float so = __shfl_xor(sE[v], off, 32);
            const float nm = fmaxf(m[v], mo);
            sE[v] = sE[v] * __expf(m[v] - nm) + so * __expf(mo - nm);
            m[v]  = nm;
            S1[v] += __shfl_xor(S1[v], off, 32);
            S2[v] += __shfl_xor(S2[v], off, 32);
            Cn[v] += __shfl_xor(Cn[v], off, 32);
        }
        if (nl == 0) {
            const int row = v + 8 * hi;
            red[wv][row][0] = m[v];
            red[wv][row][1] = sE[v];
            red[wv][row][2] = S1[v];
            red[wv][row][3] = S2[v];
            red[wv][row][4] = Cn[v];
        }
    }
    __syncthreads();

    // ---- cross-wave merge + per-row loss, then block partial
    float lossRow = 0.f;
    if (tid < 16) {
        float mm = NNEG, ss = 0.f, s1 = 0.f, s2 = 0.f, cn = 0.f;
        #pragma unroll
        for (int w8 = 0; w8 < 8; ++w8) {
            const float mo = red[w8][tid][0];
            const float so = red[w8][tid][1];
            const float nm = fmaxf(mm, mo);
            ss = ss * __expf(mm - nm) + so * __expf(mo - nm);
            mm = nm;
            s1 += red[w8][tid][2];
            s2 += red[w8][tid][3];
            cn += red[w8][tid][4];
        }
        const float lse = mm + logf(ss);
        lossRow = -(s1 - lse * s2) / cn;
    }
    if (tid < 32) {
        float lr = (tid < 16) ? lossRow : 0.f;
        for (int off = 1; off < 32; off <<= 1) lr += __shfl_xor(lr, off, 32);
        if (tid == 0) partial[blockIdx.x] = lr;
    }
}

// deterministic final reduce of 256 partials -> scalar mean
__global__ __launch_bounds__(256) void macl_fused_reduce(
    const float* __restrict__ partial, float* __restrict__ out)
{
    __shared__ float sh[8];
    const int tid = threadIdx.x;
    float v = partial[tid];
    for (int off = 1; off < 32; off <<= 1) v += __shfl_xor(v, off, 32);
    if ((tid & 31) == 0) sh[tid >> 5] = v;
    __syncthreads();
    if (tid < 8) {
        float w = sh[tid];
        for (int off = 1; off < 8; off <<= 1) w += __shfl_xor(w, off, 32);
        if (tid == 0) out[0] = w * (1.0f / 4096.0f);
    }
}

extern "C" void kernel_launch(void* const* d_in, const int* in_sizes, int n_in,
                              void* d_out, int out_size, void* d_ws, size_t ws_size,
                              hipStream_t stream) {
    (void)in_sizes; (void)n_in; (void)out_size; (void)ws_size;
    const float* feats  = (const float*)d_in[0];
    const int*   labels = (const int*)d_in[1];
    float* out     = (float*)d_out;
    float* partial = (float*)d_ws;   // 256 floats

    macl_fused_main<<<256, 256, 0, stream>>>(feats, labels, partial);
    macl_fused_reduce<<<1, 256, 0, stream>>>(partial, out);
}